// VectorSHT_75660143886825
// MI455X (gfx1250) — compile-verified
//
#include <hip/hip_runtime.h>
#include <hip/hip_bf16.h>

// ---- problem constants (match reference) ----
#define NLATC  511   // nlat == lmax
#define NLONC  512   // nlon
#define MMAXC  257   // nlon/2 + 1
#define BATCHC 8
#define NCOL   514   // 257 real cols + 257 imag cols
#define NCOLP  528   // padded to 33 tiles of 16
#define KPAD   512   // k (latitude) padded to 512 for the WMMA K loop
#define NQ     32    // 4 quantities (xr0,xi0,xr1,xi1) x 8 batch

typedef __attribute__((ext_vector_type(2))) float v2f;
typedef __attribute__((ext_vector_type(8))) float v8f;

static __device__ __forceinline__ v8f vzero8() {
  v8f z;
#pragma unroll
  for (int i = 0; i < 8; ++i) z[i] = 0.0f;
  return z;
}

// D(16x16,f32) = A(16x4,f32) * B(4x16,f32) + C
static __device__ __forceinline__ v8f wmma_f32(v2f a, v2f b, v8f c) {
  return __builtin_amdgcn_wmma_f32_16x16x4_f32(
      /*neg_a=*/false, a, /*neg_b=*/false, b,
      /*c_mod=*/(short)0, c, /*reuse_a=*/false, /*reuse_b=*/false);
}

// ---------------------------------------------------------------------------
// Kernel 0: DFT basis table tab[n][col] (col<257: cos, 257..513: -sin, rest 0)
// + zero the k=511 pad row of XF.  Angle reduced exactly via integer mod-512.
// ---------------------------------------------------------------------------
__global__ void init_tab_kernel(float* __restrict__ tab, float* __restrict__ xf) {
  int idx = blockIdx.x * blockDim.x + threadIdx.x;
  int total = NLONC * NCOLP;
  if (idx < total) {
    int n = idx / NCOLP;
    int col = idx - n * NCOLP;
    float v = 0.0f;
    if (col < NCOL) {
      int m = (col < MMAXC) ? col : (col - MMAXC);
      int r = (m * n) & (NLONC - 1);              // exact periodic reduction
      float th = (float)r * (6.2831853071795864769f / (float)NLONC);
      v = (col < MMAXC) ? __cosf(th) : (-__sinf(th));
    }
    tab[idx] = v;
  }
  if (idx < MMAXC * NQ) {              // zero XF[m][511][q] pad rows
    int m = idx / NQ;
    int q = idx - m * NQ;
    xf[((size_t)m * KPAD + (KPAD - 1)) * NQ + q] = 0.0f;
  }
}

// ---------------------------------------------------------------------------
// Kernel 1: rFFT as DFT-GEMM via WMMA, software-pipelined.
//   grid = (32 k-tiles, 16 (b,c)), block = 256 (8 waves).
//   A(16x4): 16 rings x 4 longitudes from LDS.  B(4x16): basis (L2-hot).
//   Wave owns tiles t = wave+8i: i<4 unconditional, 5th tile only on wave 0.
// ---------------------------------------------------------------------------
__global__ __launch_bounds__(256) void dft_kernel(const float* __restrict__ x,
                                                  const float* __restrict__ tab,
                                                  float* __restrict__ xf) {
  __shared__ float xlds[16 * NLONC];        // 32 KB
  const int kt   = blockIdx.x;              // 0..31
  const int bc   = blockIdx.y;              // 0..15  (= b*2 + c)
  const int b    = bc >> 1;
  const int c    = bc & 1;
  const int tid  = threadIdx.x;
  const int wave = __builtin_amdgcn_readfirstlane(tid >> 5);  // scalar
  const int lane = tid & 31;
  const bool has5 = (wave == 0);            // only wave 0 owns tile 32

  // cooperatively stage 16 rings x 512 longitudes (clamp ring index)
  const float* xrow = x + (size_t)bc * NLATC * NLONC;
  for (int i = tid; i < 16 * NLONC; i += 256) {
    int kr = i >> 9;
    int n  = i & (NLONC - 1);
    int k  = kt * 16 + kr;
    if (k > NLATC - 1) k = NLATC - 1;
    xlds[i] = xrow[(size_t)k * NLONC + n];
  }
  __syncthreads();

  const int r0   = (lane < 16) ? 0 : 2;     // K split across lane halves
  const int arow = lane & 15;
  int colof[5];
#pragma unroll
  for (int i = 0; i < 5; ++i) colof[i] = (wave + i * 8) * 16 + (lane & 15);

  v8f acc[5];
#pragma unroll
  for (int i = 0; i < 5; ++i) acc[i] = vzero8();

  // --- prologue: load K-step 0 ---
  v2f a_c, b_c[5];
  a_c.x = xlds[arow * NLONC + r0];
  a_c.y = xlds[arow * NLONC + r0 + 1];
#pragma unroll
  for (int i = 0; i < 4; ++i) {
    b_c[i].x = tab[r0 * NCOLP + colof[i]];
    b_c[i].y = tab[(r0 + 1) * NCOLP + colof[i]];
  }
  if (has5) {
    b_c[4].x = tab[r0 * NCOLP + colof[4]];
    b_c[4].y = tab[(r0 + 1) * NCOLP + colof[4]];
  }

  for (int n0 = 0; n0 < NLONC; n0 += 4) {
    // issue next K-step's loads (last iteration reloads step 0; discarded)
    int nn = (n0 + 4 < NLONC) ? (n0 + 4) : 0;
    v2f a_n, b_n[5];
    a_n.x = xlds[arow * NLONC + nn + r0];
    a_n.y = xlds[arow * NLONC + nn + r0 + 1];
#pragma unroll
    for (int i = 0; i < 4; ++i) {
      b_n[i].x = tab[(nn + r0) * NCOLP + colof[i]];
      b_n[i].y = tab[(nn + r0 + 1) * NCOLP + colof[i]];
    }
    if (has5) {
      b_n[4].x = tab[(nn + r0) * NCOLP + colof[4]];
      b_n[4].y = tab[(nn + r0 + 1) * NCOLP + colof[4]];
    }

    // WMMAs consume the previous iteration's registers
#pragma unroll
    for (int i = 0; i < 4; ++i) acc[i] = wmma_f32(a_c, b_c[i], acc[i]);
    if (has5) acc[4] = wmma_f32(a_c, b_c[4], acc[4]);

    a_c = a_n;
#pragma unroll
    for (int i = 0; i < 4; ++i) b_c[i] = b_n[i];
    if (has5) b_c[4] = b_n[4];
  }

  // scatter D tiles into XF[m][k][q]; tiles i<4 have col<=511 (no guard)
#pragma unroll
  for (int i = 0; i < 5; ++i) {
    if (i == 4 && !has5) continue;          // scalar
    int col = colof[i];
    if (i == 4 && col >= NCOL) continue;    // lane-divergent, stores only
    int isIm = (col >= MMAXC) ? 1 : 0;
    int m    = isIm ? (col - MMAXC) : col;
    int q    = (c * 2 + isIm) * 8 + b;
#pragma unroll
    for (int v = 0; v < 8; ++v) {
      int M = v + ((lane < 16) ? 0 : 8);
      int k = kt * 16 + M;
      if (k <= NLATC - 1) {
        xf[((size_t)m * KPAD + k) * NQ + q] = acc[i][v];
      }
    }
  }
}

// ---------------------------------------------------------------------------
// Kernel 2: per-m Legendre contraction via WMMA + recombination.
//   grid = (257 m, 4 l-groups), block = 256 (8 waves); wave owns one 16-l tile.
//   A(16x4) = w[c][m][l][k] tile (contiguous along k; streamed ONCE from HBM).
//   B(4x16) = XF[m][k][q] tile (64 KB slab per m, L2-hot).
//   Software-pipelined: next K-step's 8 loads issue before this step's WMMAs.
// ---------------------------------------------------------------------------
__global__ __launch_bounds__(256) void leg_kernel(const float* __restrict__ w,
                                                  const float* __restrict__ xf,
                                                  float* __restrict__ out) {
  __shared__ float sc[8 * 4 * 32 * 8];      // 8 waves x 4 tiles x 256 f32 = 32 KB
  const int m    = blockIdx.x;              // 0..256
  const int lgrp = blockIdx.y;              // 0..3
  const int tid  = threadIdx.x;
  const int wave = tid >> 5;
  const int lane = tid & 31;
  const int lt   = lgrp * 8 + wave;         // 0..31 (l tile)
  const int r0   = (lane < 16) ? 0 : 2;
  const int lrow = lane & 15;

  int l_a = lt * 16 + lrow;
  if (l_a > NLATC - 1) l_a = NLATC - 1;     // clamp pad row (discarded later)

  const size_t wcomp = (size_t)MMAXC * NLATC * NLATC;
  const float* a0base = w + ((size_t)m * NLATC + l_a) * NLATC;   // w0[m][l][k]
  const float* a1base = a0base + wcomp;                          // w1[m][l][k]
  const float* bbase  = xf + (size_t)m * KPAD * NQ;

  const int q0 = lane & 15;
  const int q1 = 16 + (lane & 15);

  v8f acc00 = vzero8(), acc01 = vzero8(), acc10 = vzero8(), acc11 = vzero8();

  // --- prologue: load K-step 0 ---
  v2f a0c, a1c, b0c, b1c;
  {
    int ka = r0, kb = r0 + 1;
    a0c.x = a0base[ka];  a0c.y = a0base[kb];
    a1c.x = a1base[ka];  a1c.y = a1base[kb];
    b0c.x = bbase[(size_t)ka * NQ + q0];  b0c.y = bbase[(size_t)kb * NQ + q0];
    b1c.x = bbase[(size_t)ka * NQ + q1];  b1c.y = bbase[(size_t)kb * NQ + q1];
  }

  for (int k0 = 0; k0 < KPAD; k0 += 4) {
    if ((k0 & 63) == 0) {                   // prefetch the weight stream
      __builtin_prefetch(a0base + k0 + 128, 0, 0);
      __builtin_prefetch(a1base + k0 + 128, 0, 0);
    }
    // issue next K-step's loads (last iteration reloads step 0; discarded)
    int kn = (k0 + 4 < KPAD) ? (k0 + 4) : 0;
    int ka = kn + r0;
    int kb = ka + 1;
    int kbc = (kb > NLATC - 1) ? (NLATC - 1) : kb;  // clamp addr; B row 511 = 0
    v2f a0n, a1n, b0n, b1n;
    a0n.x = a0base[ka];  a0n.y = a0base[kbc];
    a1n.x = a1base[ka];  a1n.y = a1base[kbc];
    b0n.x = bbase[(size_t)ka * NQ + q0];  b0n.y = bbase[(size_t)kb * NQ + q0];
    b1n.x = bbase[(size_t)ka * NQ + q1];  b1n.y = bbase[(size_t)kb * NQ + q1];

    // WMMAs consume the previous iteration's registers (loads in flight above)
    acc00 = wmma_f32(a0c, b0c, acc00);
    acc01 = wmma_f32(a0c, b1c, acc01);
    acc10 = wmma_f32(a1c, b0c, acc10);
    acc11 = wmma_f32(a1c, b1c, acc11);

    a0c = a0n; a1c = a1n; b0c = b0n; b1c = b1n;
  }

  // spill the 4 D tiles to this wave's LDS scratch (cross-lane recombination)
  float* mysc = sc + wave * (4 * 32 * 8);
#pragma unroll
  for (int v = 0; v < 8; ++v) {
    mysc[(0 * 32 + lane) * 8 + v] = acc00[v];
    mysc[(1 * 32 + lane) * 8 + v] = acc01[v];
    mysc[(2 * 32 + lane) * 8 + v] = acc10[v];
    mysc[(3 * 32 + lane) * 8 + v] = acc11[v];
  }

  // each lane recombines 4 (l,b) pairs into the 2 complex outputs
  float2* o = (float2*)out;
#pragma unroll
  for (int p = 0; p < 4; ++p) {
    int pid = lane + 32 * p;                // 0..127
    int ll  = pid >> 3;                     // 0..15 local l
    int bb  = pid & 7;                      // batch
    int l   = lt * 16 + ll;
    if (l >= NLATC) continue;
    float g[2][4];                          // g[wc][quant]
#pragma unroll
    for (int wc = 0; wc < 2; ++wc) {
#pragma unroll
      for (int qn = 0; qn < 4; ++qn) {
        int tile  = wc * 2 + (qn >> 1);
        int coln  = (qn & 1) * 8 + bb;
        int laned = ((ll >= 8) ? 16 : 0) + coln;
        int v     = ll & 7;
        g[wc][qn] = mysc[(tile * 32 + laned) * 8 + v];
      }
    }
    // o0 = (leg(xr0,w0)-leg(xi1,w1)) + i(leg(xi0,w0)+leg(xr1,w1))
    // o1 = (-leg(xi0,w1)-leg(xr1,w0)) + i(leg(xr0,w1)-leg(xi1,w0))
    float o0re =  g[0][0] - g[1][3];
    float o0im =  g[0][1] + g[1][2];
    float o1re = -g[1][1] - g[0][2];
    float o1im =  g[1][0] - g[0][3];
    size_t base0 = ((size_t)(bb * 2 + 0) * NLATC + l) * MMAXC + m;
    size_t base1 = ((size_t)(bb * 2 + 1) * NLATC + l) * MMAXC + m;
    o[base0] = make_float2(o0re, o0im);
    o[base1] = make_float2(o1re, o1im);
  }
}

// ---------------------------------------------------------------------------
extern "C" void kernel_launch(void* const* d_in, const int* in_sizes, int n_in,
                              void* d_out, int out_size, void* d_ws, size_t ws_size,
                              hipStream_t stream) {
  (void)in_sizes; (void)n_in; (void)out_size; (void)ws_size;
  const float* x = (const float*)d_in[0];    // [8, 2, 511, 512]
  const float* w = (const float*)d_in[1];    // [2, 257, 511, 511]
  float* out = (float*)d_out;                // [8, 2, 511, 257] complex64

  float* tab = (float*)d_ws;                 // [512][528]  = 1.06 MB
  float* xf  = tab + (size_t)NLONC * NCOLP;  // [257][512][32] = 16.8 MB

  int initN = NLONC * NCOLP;
  init_tab_kernel<<<(initN + 255) / 256, 256, 0, stream>>>(tab, xf);

  dim3 g1(32, 16);                           // k-tiles x (b,c)
  dft_kernel<<<g1, 256, 0, stream>>>(x, tab, xf);

  dim3 g2(MMAXC, 4);                         // m x l-groups
  leg_kernel<<<g2, 256, 0, stream>>>(w, xf, out);
}